// ModelNew_66924180406830
// MI455X (gfx1250) — compile-verified
//
#include <hip/hip_runtime.h>

typedef float v2f __attribute__((ext_vector_type(2)));
typedef float v8f __attribute__((ext_vector_type(8)));

#define N_IMG 128
#define C_IN 3
#define H_IN 384
#define W_IN 384
#define C_OUT 16
#define H_OUT 382
#define W_OUT 382
#define K_RED 27          // 3*3*3 reduction, padded to 28 (7 x K=4 WMMA steps)
#define W_TILES 24        // ceil(382/16); last tile shifted to w0=366 (full tile)
#define WAVES_PER_BLOCK 8
#define SUB_TOT 0.7f      // SUB1 + SUB2
#define L2E 1.44269504088896f

// OFF[k] = c*H*W + r*W + s for k = c*9 + r*3 + s ; OFF[27] aliases OFF[26]
// (the K=27 pad lane loads a valid address; its B value is zeroed instead).
__device__ constexpr int OFFX[28] = {
    0,      1,      2,      384,    385,    386,    768,    769,    770,
    147456, 147457, 147458, 147840, 147841, 147842, 148224, 148225, 148226,
    294912, 294913, 294914, 295296, 295297, 295298, 295680, 295681, 295682,
    295682
};

// One wave -> full 16(pixels-along-W) x 16(out-channels) fp32 output tile via
// 7 chained V_WMMA_F32_16X16X4_F32, then bias + Mish, 4x b64 stores per lane.
// All tiles are full (last W tile is shifted/overlapped), so there is no
// bounds logic anywhere: EXEC stays all-ones, no divergent branches.
__global__ __launch_bounds__(256) void conv3x3_mish_wmma(
    const float* __restrict__ x,     // (128,3,384,384)
    const float* __restrict__ wgt,   // (16,3,3,3) -> flat [o*27 + k], 432 floats
    const float* __restrict__ bias,  // (16,)
    float* __restrict__ out)         // (128,16,382,382)
{
    const int lane = threadIdx.x & 31;
    const int half = lane >> 4;      // K-half selector (fp32 WMMA A/B layout)
    const int mrow = lane & 15;      // A: pixel row M ; B/D: channel column N

    // ---- wave-uniform tile coordinates, forced into SGPRs ----
    const int wid = __builtin_amdgcn_readfirstlane(
        blockIdx.x * WAVES_PER_BLOCK + (threadIdx.x >> 5));
    const int tiles_per_img = H_OUT * W_TILES;
    const int n   = wid / tiles_per_img;
    const int rem = wid - n * tiles_per_img;
    const int h   = rem / W_TILES;
    int w0        = (rem - h * W_TILES) * 16;
    if (w0 > W_OUT - 16) w0 = W_OUT - 16;          // 366: full tile, overlaps prev

    // uniform base pointers (SGPR); per-lane data reached via small voffsets
    const float* __restrict__ xs =
        x + ((long)n * C_IN * H_IN * W_IN + (long)h * W_IN + w0);
    float* __restrict__ os =
        out + (((long)n * C_OUT) * H_OUT + h) * (long)W_OUT + w0;

    // per-lane weight bases (immediate-offset loads thereafter)
    const float* __restrict__ wb = wgt + (mrow * K_RED + 2 * half);
    // clamped base for the single K=27 pad load (lane mrow=15,half=1 only)
    int wyo = mrow * K_RED + 2 * half;
    if (wyo > 406) wyo = 406;                       // keeps wb2[25] <= 431
    const float* __restrict__ wb2 = wgt + wyo;

    v8f acc = {0.f, 0.f, 0.f, 0.f, 0.f, 0.f, 0.f, 0.f};

    // ---- K loop: 7 chunks of K=4 ----
    #pragma unroll
    for (int chunk = 0; chunk < 7; ++chunk) {
        const int k0 = 4 * chunk;
        // A(16x4): lane half 0 -> K = k0,k0+1 ; half 1 -> K = k0+2,k0+3
        const int o0 = (half ? OFFX[k0 + 2] : OFFX[k0 + 0]) + mrow;
        const int o1 = (half ? OFFX[k0 + 3] : OFFX[k0 + 1]) + mrow;
        v2f a, b;
        a.x = xs[o0];
        a.y = xs[o1];
        // B(4x16): same K split; wgt is K-contiguous -> immediate offsets
        b.x = wb[k0];
        if (chunk < 6) {
            b.y = wb[k0 + 1];
        } else {
            b.y = wb2[k0 + 1];                      // k=25 (half0) / pad (half1)
            b.y = half ? 0.f : b.y;                 // zero the K=27 pad column
        }
        acc = __builtin_amdgcn_wmma_f32_16x16x4_f32(
                  false, a, false, b, (short)0, acc, false, false);
    }

    // ---- epilogue: t = conv + bias - 0.7 ; mish(t) = t*u*(u+2)/(u*(u+2)+2) ----
    const float bvadj = bias[mrow] - SUB_TOT;       // D layout: N = lane&15
    // per-lane output offset: channel plane + M half (pixels r / r+8)
    const int ovoff = mrow * (H_OUT * W_OUT) + 8 * half;
    float* __restrict__ op = os + ovoff;            // 8B-aligned for even r

    #pragma unroll
    for (int p = 0; p < 4; ++p) {
        v2f t, u, num, d, m;
        t.x = acc[2 * p]     + bvadj;
        t.y = acc[2 * p + 1] + bvadj;
        u.x = __builtin_amdgcn_exp2f(fminf(t.x, 30.f) * L2E);   // e^t
        u.y = __builtin_amdgcn_exp2f(fminf(t.y, 30.f) * L2E);
        num = u * (u + 2.f);                        // packed-f32 friendly
        d   = num + 2.f;
        v2f r; r.x = __builtin_amdgcn_rcpf(d.x); r.y = __builtin_amdgcn_rcpf(d.y);
        m   = (t * num) * r;
        *(v2f*)(op + 2 * p) = m;                    // global_store_b64
    }
}

extern "C" void kernel_launch(void* const* d_in, const int* in_sizes, int n_in,
                              void* d_out, int out_size, void* d_ws, size_t ws_size,
                              hipStream_t stream) {
    const float* x    = (const float*)d_in[0];
    const float* wgt  = (const float*)d_in[1];
    const float* bias = (const float*)d_in[2];
    float* out        = (float*)d_out;

    const int total_waves = N_IMG * H_OUT * W_TILES;                 // 1,173,504
    const int blocks = (total_waves + WAVES_PER_BLOCK - 1) / WAVES_PER_BLOCK; // 146,688
    conv3x3_mish_wmma<<<blocks, 256, 0, stream>>>(x, wgt, bias, out);
}